// CausalSelfAttention_12824772346099
// MI455X (gfx1250) — compile-verified
//
#include <hip/hip_runtime.h>
#include <hip/hip_bf16.h>

typedef float v2f __attribute__((ext_vector_type(2)));
typedef float v8f __attribute__((ext_vector_type(8)));

// ---------------------------------------------------------------------------
// Generic NT GEMM with bias:  C[M x N] = A[M x K] * B[N x K]^T + bias[N]
// A row-major (lda = K), B row-major (ldb = K), C row-major (ldc = N).
// Block = 128 threads (4 waves). Block tile 64x64, K-chunk 32, double-buffered
// LDS staged with CDNA5 async global->LDS (ASYNCcnt), fp32 WMMA 16x16x4.
// Requires: M%64==0, N%64==0, K%32==0 (true for all calls here).
// ---------------------------------------------------------------------------
#define GPAD 36  // LDS row pad (floats): 144B rows -> 16B aligned, no bank conflicts

__global__ __launch_bounds__(128) void gemm_nt_bias_f32(
    const float* __restrict__ A, const float* __restrict__ B,
    const float* __restrict__ bias, float* __restrict__ C,
    int M, int N, int K) {
  __shared__ __align__(16) float As[2][64 * GPAD];
  __shared__ __align__(16) float Bs[2][64 * GPAD];

  const int tid  = threadIdx.x;
  const int wave = tid >> 5;
  const int lane = tid & 31;
  const int lo   = lane & 15;
  const int hi   = lane >> 4;

  const int bN = blockIdx.x * 64;
  const int bM = blockIdx.y * 64;

  // Async-stage one 64x32 chunk of A and B into LDS buffer `b`.
  // 512 float4 per matrix, 128 threads -> 4 async b128 ops each per matrix.
  auto stage = [&](int k0, int b) {
#pragma unroll
    for (int i = 0; i < 4; ++i) {
      int idx = tid + i * 128;       // 0..511
      int rr  = idx >> 3;            // row in tile
      int cc  = (idx & 7) << 2;      // col (float4 granularity)
      unsigned la = (unsigned)(uintptr_t)&As[b][rr * GPAD + cc];
      const float* ga = &A[(size_t)(bM + rr) * K + k0 + cc];
      asm volatile("global_load_async_to_lds_b128 %0, %1, off"
                   :: "v"(la), "v"(ga) : "memory");
      unsigned lb = (unsigned)(uintptr_t)&Bs[b][rr * GPAD + cc];
      const float* gb = &B[(size_t)(bN + rr) * K + k0 + cc];
      asm volatile("global_load_async_to_lds_b128 %0, %1, off"
                   :: "v"(lb), "v"(gb) : "memory");
    }
  };

  v8f acc[4] = {};

  stage(0, 0);
  int cur = 0;
  for (int k0 = 0; k0 < K; k0 += 32) {
    // Wait for this wave's async transfers, then block-wide data-ready barrier.
    asm volatile("s_wait_asynccnt 0" ::: "memory");
    __syncthreads();
    // Prefetch the next chunk into the other buffer while we compute.
    if (k0 + 32 < K) stage(k0 + 32, cur ^ 1);

    const float* as = &As[cur][0];
    const float* bs = &Bs[cur][0];
#pragma unroll
    for (int kk = 0; kk < 32; kk += 4) {
      // A operand (16x4 fp32): lanes 0-15 -> K=kk,kk+1 ; lanes 16-31 -> kk+2,kk+3
      v2f a = *(const v2f*)&as[(16 * wave + lo) * GPAD + kk + 2 * hi];
#pragma unroll
      for (int ns = 0; ns < 4; ++ns) {
        v2f b = *(const v2f*)&bs[(16 * ns + lo) * GPAD + kk + 2 * hi];
        acc[ns] = __builtin_amdgcn_wmma_f32_16x16x4_f32(
            false, a, false, b, (short)0, acc[ns], false, false);
      }
    }
    cur ^= 1;
  }

  // Epilogue: C/D layout -> element (m = v + 8*hi, n = lo) per 16x16 sub-tile
  const int m0 = bM + 16 * wave;
#pragma unroll
  for (int ns = 0; ns < 4; ++ns) {
    int col  = bN + 16 * ns + lo;
    float bv = bias[col];
#pragma unroll
    for (int v = 0; v < 8; ++v) {
      int row = m0 + v + 8 * hi;
      C[(size_t)row * N + col] = acc[ns][v] + bv;
    }
  }
}

// ---------------------------------------------------------------------------
// Flash-style causal attention, fp32 WMMA.
// qkv: (2048 x 3072), row = [q(1024) | k(1024) | v(1024)], head h -> cols h*64..h*64+63
// One wave handles one (head, 16-row query tile). Online softmax in registers
// (lanes 0-15 own one row each); P round-trips through wave-private LDS to
// convert C/D layout -> A layout for the P*V WMMA.
// ---------------------------------------------------------------------------
#define LDQ 3072
#define SPAD 18  // St row pad (floats): 72B rows -> 8B aligned v2f reads

__global__ __launch_bounds__(128) void attn_causal_f32(
    const float* __restrict__ qkv, float* __restrict__ out) {
  // per-wave region: St[16*18] + Al[16] + Ll[16] = 320 floats
  __shared__ __align__(16) float lds[4 * 320];

  const int tid  = threadIdx.x;
  const int wave = tid >> 5;
  const int lane = tid & 31;
  const int lo   = lane & 15;
  const int hi   = lane >> 4;

  float* St = &lds[wave * 320];
  float* Al = St + 288;
  float* Ll = St + 304;

  const int unit = blockIdx.x * 4 + wave;  // 0..2047
  const int h    = unit >> 7;              // head 0..15
  const int qt   = unit & 127;             // q tile 0..127
  const int q0   = qt * 16;

  // Preload Q tile (16x64) in A layout, pre-scaled by 1/sqrt(D)
  const float scale = 0.125f;
  v2f qa[16];
  {
    const float* qrow = qkv + (size_t)(q0 + lo) * LDQ + h * 64 + 2 * hi;
#pragma unroll
    for (int c = 0; c < 16; ++c) {
      v2f t = *(const v2f*)&qrow[4 * c];
      qa[c] = t * scale;
    }
  }

  v8f o[4] = {};
  float mreg = -1e30f;  // running row max (lanes 0-15 own row = lo)
  float lreg = 0.0f;    // running row sum

  for (int kt = 0; kt <= qt; ++kt) {
    const int k0 = kt * 16;

    // ---- S = (Q*scale) K^T : 16 WMMAs over D=64 ----
    v8f s = {};
    {
      const float* krow = qkv + (size_t)(k0 + lo) * LDQ + 1024 + h * 64 + 2 * hi;
#pragma unroll
      for (int c = 0; c < 16; ++c) {
        v2f kb = *(const v2f*)&krow[4 * c];
        s = __builtin_amdgcn_wmma_f32_16x16x4_f32(
            false, qa[c], false, kb, (short)0, s, false, false);
      }
    }

    // ---- causal mask on the diagonal tile ----
    if (kt == qt) {
#pragma unroll
      for (int v = 0; v < 8; ++v) {
        int m = v + 8 * hi;
        if (lo > m) s[v] = -1e30f;
      }
    }

    // ---- spill S tile to wave-private LDS ----
#pragma unroll
    for (int v = 0; v < 8; ++v) St[(v + 8 * hi) * SPAD + lo] = s[v];
    asm volatile("s_wait_dscnt 0" ::: "memory");

    // ---- online softmax: lanes 0-15 each own one row ----
    if (lane < 16) {
      const int r = lo;
      float rmax = mreg;
#pragma unroll
      for (int j = 0; j < 16; ++j) rmax = fmaxf(rmax, St[r * SPAD + j]);
      float alpha = __expf(mreg - rmax);
      float rsum  = 0.0f;
#pragma unroll
      for (int j = 0; j < 16; ++j) {
        float p = __expf(St[r * SPAD + j] - rmax);
        St[r * SPAD + j] = p;
        rsum += p;
      }
      Al[r] = alpha;
      mreg  = rmax;
      lreg  = lreg * alpha + rsum;
    }
    asm volatile("s_wait_dscnt 0" ::: "memory");

    // ---- rescale O accumulators by per-row alpha ----
#pragma unroll
    for (int v = 0; v < 8; ++v) {
      float av = Al[v + 8 * hi];
      o[0][v] *= av; o[1][v] *= av; o[2][v] *= av; o[3][v] *= av;
    }

    // ---- O += P * V : 4 K-chunks x 4 D-subtiles = 16 WMMAs ----
    const float* vbase = qkv + 2048 + h * 64;
#pragma unroll
    for (int c = 0; c < 4; ++c) {
      v2f pa = *(const v2f*)&St[lo * SPAD + 4 * c + 2 * hi];
      const int kr = k0 + 4 * c + 2 * hi;
      const float* vr = vbase + (size_t)kr * LDQ + lo;
#pragma unroll
      for (int ns = 0; ns < 4; ++ns) {
        v2f b;
        b.x = vr[16 * ns];
        b.y = vr[16 * ns + LDQ];
        o[ns] = __builtin_amdgcn_wmma_f32_16x16x4_f32(
            false, pa, false, b, (short)0, o[ns], false, false);
      }
    }
  }

  // ---- finalize: divide by row sums, write (S, H, D) packed as (S, E) ----
  if (lane < 16) Ll[lo] = lreg;
  asm volatile("s_wait_dscnt 0" ::: "memory");
#pragma unroll
  for (int v = 0; v < 8; ++v) {
    float inv   = 1.0f / Ll[v + 8 * hi];
    int row     = q0 + v + 8 * hi;
    size_t base = (size_t)row * 1024 + h * 64 + lo;
    out[base +  0] = o[0][v] * inv;
    out[base + 16] = o[1][v] * inv;
    out[base + 32] = o[2][v] * inv;
    out[base + 48] = o[3][v] * inv;
  }
}

// ---------------------------------------------------------------------------
// Launch: qkv GEMM -> causal attention -> proj GEMM (stream-ordered)
// ---------------------------------------------------------------------------
extern "C" void kernel_launch(void* const* d_in, const int* in_sizes, int n_in,
                              void* d_out, int out_size, void* d_ws, size_t ws_size,
                              hipStream_t stream) {
  (void)in_sizes; (void)n_in; (void)out_size; (void)ws_size;

  const float* x      = (const float*)d_in[0];  // (2048, 1024)
  const float* w_attn = (const float*)d_in[1];  // (3072, 1024)
  const float* b_attn = (const float*)d_in[2];  // (3072,)
  const float* w_proj = (const float*)d_in[3];  // (1024, 1024)
  const float* b_proj = (const float*)d_in[4];  // (1024,)
  float* out = (float*)d_out;                   // (2048, 1024)

  float* qkv      = (float*)d_ws;                       // 2048*3072 f32 = 24 MB
  float* attn_out = qkv + (size_t)2048 * 3072;          // 2048*1024 f32 =  8 MB

  dim3 blk(128);

  // qkv = x @ w_attn^T + b_attn   (M=2048, N=3072, K=1024)
  gemm_nt_bias_f32<<<dim3(3072 / 64, 2048 / 64), blk, 0, stream>>>(
      x, w_attn, b_attn, qkv, 2048, 3072, 1024);

  // attn_out = softmax(causal(QK^T/sqrt(D))) V   (2048 wave-units)
  attn_causal_f32<<<dim3(512), blk, 0, stream>>>(qkv, attn_out);

  // out = attn_out @ w_proj^T + b_proj   (M=2048, N=1024, K=1024)
  gemm_nt_bias_f32<<<dim3(1024 / 64, 2048 / 64), blk, 0, stream>>>(
      attn_out, w_proj, b_proj, out, 2048, 1024, 1024);
}